// MLPEnsemble_78288663871542
// MI455X (gfx1250) — compile-verified
//
#include <hip/hip_runtime.h>
#include <hip/hip_bf16.h>
#include <math.h>

typedef __attribute__((ext_vector_type(2))) float v2f;
typedef __attribute__((ext_vector_type(4))) float v4f;
typedef __attribute__((ext_vector_type(8))) float v8f;
typedef __attribute__((ext_vector_type(4))) int   v4i;

#define BATCH 8192
#define INSZ  1024
#define HID   2048
#define NEXP  30

#define MT 256        // rows per workgroup (8 waves x 32 rows)
#define NT 128        // H columns per outer iteration (full width per wave)
#define KC 32         // K chunk staged in LDS

#define XS_STRIDE 36   // padded: conflict-free A-frag reads, 16B-aligned rows
#define WS_STRIDE 144  // padded: conflict-free B-frag reads, 16B-aligned rows

#if __has_builtin(__builtin_amdgcn_global_load_async_to_lds_b128)
#define HAVE_ASYNC_LDS 1
#else
#define HAVE_ASYNC_LDS 0
#endif

__device__ __forceinline__ void wait_async_zero()
{
#if HAVE_ASYNC_LDS
#if __has_builtin(__builtin_amdgcn_s_wait_asynccnt)
    __builtin_amdgcn_s_wait_asynccnt(0);
#else
    asm volatile("s_wait_asynccnt 0x0" ::: "memory");
#endif
#endif
}

__device__ __forceinline__ void async_copy16(const float* g, float* l)
{
#if HAVE_ASYNC_LDS
    __builtin_amdgcn_global_load_async_to_lds_b128(
        (__attribute__((address_space(1))) v4i*)g,
        (__attribute__((address_space(3))) v4i*)l, 0, 0);
#else
    *(v4f*)l = *(const v4f*)g;
#endif
}

// Stage X[m0:m0+256, k0:k0+32] and W1e[k0:k0+32, h0:h0+128] into LDS buffers.
// Async path: GLOBAL_LOAD_ASYNC_TO_LDS_B128 (ASYNCcnt), no VGPR round-trip.
__device__ __forceinline__ void stage_chunk(const float* __restrict__ x,
                                            const float* __restrict__ W1e,
                                            int m0, int h0, int k0,
                                            float* __restrict__ Xbuf,
                                            float* __restrict__ Wbuf,
                                            int tid)
{
#pragma unroll
    for (int i = 0; i < 8; ++i) {
        int idx = tid + 256 * i;
        int row = idx >> 3;       // 8 float4 per X row, 256 rows
        int c4  = idx & 7;
        async_copy16(x + (size_t)(m0 + row) * INSZ + k0 + c4 * 4,
                     &Xbuf[row * XS_STRIDE + c4 * 4]);
    }
#pragma unroll
    for (int i = 0; i < 4; ++i) {
        int idx  = tid + 256 * i;
        int krow = idx >> 5;      // 32 float4 per W row
        int c4   = idx & 31;
        async_copy16(W1e + (size_t)(k0 + krow) * HID + h0 + c4 * 4,
                     &Wbuf[krow * WS_STRIDE + c4 * 4]);
    }
}

__global__ __launch_bounds__(256)
void moe_fc_fused_kernel(const float* __restrict__ x,
                         const float* __restrict__ W1,
                         const float* __restrict__ b1,
                         const float* __restrict__ W2,
                         const float* __restrict__ b2,
                         float* __restrict__ logits)
{
    __shared__ float XsB[2][MT * XS_STRIDE];   // ping-pong X tiles (2 x 36 KB)
    __shared__ float WsB[2][KC * WS_STRIDE];   // ping-pong W1 tiles (2 x 18 KB)

    const int e    = blockIdx.y;
    const int m0   = blockIdx.x * MT;
    const int tid  = threadIdx.x;
    const int lane = tid & 31;
    const int wave_m = tid >> 5;  // 0..7 -> which 32-row slab (wave owns full NT)
    const int l16    = lane & 15;
    const int lhalf  = lane >> 4;

    const float* __restrict__ W1e = W1 + (size_t)e * INSZ * HID;

    float racc[2][8];
#pragma unroll
    for (int mt = 0; mt < 2; ++mt)
#pragma unroll
        for (int v = 0; v < 8; ++v) racc[mt][v] = 0.0f;

    // ---- software pipeline prologue: issue chunk (h0=0, k0=0) into buffer 0
    stage_chunk(x, W1e, m0, 0, 0, XsB[0], WsB[0], tid);
    int cbuf = 0;

    for (int h0 = 0; h0 < HID; h0 += NT) {
        v8f acc[2][8];
#pragma unroll
        for (int mt = 0; mt < 2; ++mt)
#pragma unroll
            for (int nt = 0; nt < 8; ++nt) {
                v8f z = {0.f, 0.f, 0.f, 0.f, 0.f, 0.f, 0.f, 0.f};
                acc[mt][nt] = z;
            }

        for (int k0 = 0; k0 < INSZ; k0 += KC) {
            // current chunk was issued one iteration ago: drain and publish it
            wait_async_zero();
            __syncthreads();

            // issue the next chunk into the other buffer (overlaps the WMMAs)
            int nk = k0 + KC, nh = h0;
            if (nk == INSZ) { nk = 0; nh = h0 + NT; }
            if (nh < HID)
                stage_chunk(x, W1e, m0, nh, nk, XsB[cbuf ^ 1], WsB[cbuf ^ 1], tid);

            const float* __restrict__ Xs = XsB[cbuf];
            const float* __restrict__ Ws = WsB[cbuf];

#pragma unroll
            for (int kk = 0; kk < KC; kk += 4) {
                const int kf = kk + lhalf * 2;
                v2f a[2];
#pragma unroll
                for (int mt = 0; mt < 2; ++mt) {
                    int m = wave_m * 32 + mt * 16 + l16;
                    a[mt] = *(const v2f*)(&Xs[m * XS_STRIDE + kf]);
                }
                v2f bf[8];
#pragma unroll
                for (int nt = 0; nt < 8; ++nt) {
                    int n = nt * 16 + l16;
                    bf[nt].x = Ws[(kf + 0) * WS_STRIDE + n];
                    bf[nt].y = Ws[(kf + 1) * WS_STRIDE + n];
                }
#pragma unroll
                for (int mt = 0; mt < 2; ++mt)
#pragma unroll
                    for (int nt = 0; nt < 8; ++nt)
                        acc[mt][nt] = __builtin_amdgcn_wmma_f32_16x16x4_f32(
                            false, a[mt], false, bf[nt],
                            (short)0, acc[mt][nt], false, false);
            }
            cbuf ^= 1;
        }

        // ---- fused epilogue for this H tile: +b1, ReLU, *W2, accumulate
#pragma unroll
        for (int nt = 0; nt < 8; ++nt) {
            int n = h0 + nt * 16 + l16;
            float b1v = b1[(size_t)e * HID + n];
            float w2v = W2[(size_t)e * HID + n];
#pragma unroll
            for (int mt = 0; mt < 2; ++mt) {
#pragma unroll
                for (int v = 0; v < 8; ++v) {
                    float hv = acc[mt][nt][v] + b1v;
                    hv = hv > 0.0f ? hv : 0.0f;
                    racc[mt][v] += hv * w2v;
                }
            }
        }
    }

    // ---- reduce over N: 16 lanes per half hold different n for the same row.
    // Each wave owns the full NT width, so rows are wave-exclusive: store direct.
    const float b2e = b2[e];
#pragma unroll
    for (int mt = 0; mt < 2; ++mt) {
#pragma unroll
        for (int v = 0; v < 8; ++v) {
            float r = racc[mt][v];
            r += __shfl_xor(r, 1);
            r += __shfl_xor(r, 2);
            r += __shfl_xor(r, 4);
            r += __shfl_xor(r, 8);
            if (l16 == 0) {
                int row = wave_m * 32 + mt * 16 + v + 8 * lhalf;
                logits[(size_t)e * BATCH + m0 + row] = r + b2e;
            }
        }
    }
}

__global__ __launch_bounds__(256)
void moe_sigmoid_mean_kernel(const float* __restrict__ logits,
                             float* __restrict__ out)
{
    int b = blockIdx.x * blockDim.x + threadIdx.x;
    if (b >= BATCH) return;
    float s = 0.0f;
#pragma unroll
    for (int e = 0; e < NEXP; ++e) {
        float z = logits[(size_t)e * BATCH + b];
        s += 1.0f / (1.0f + expf(-z));
    }
    out[b] = s * (1.0f / NEXP);
}

extern "C" void kernel_launch(void* const* d_in, const int* in_sizes, int n_in,
                              void* d_out, int out_size, void* d_ws, size_t ws_size,
                              hipStream_t stream)
{
    const float* x  = (const float*)d_in[0];
    const float* W1 = (const float*)d_in[1];
    const float* b1 = (const float*)d_in[2];
    const float* W2 = (const float*)d_in[3];
    const float* b2 = (const float*)d_in[4];
    float* out    = (float*)d_out;
    float* logits = (float*)d_ws;   // NEXP * BATCH floats < 1 MB

    dim3 grid1(BATCH / MT, NEXP);   // x fastest -> same-expert WGs adjacent (W1_e L2 reuse)
    moe_fc_fused_kernel<<<grid1, 256, 0, stream>>>(x, W1, b1, W2, b2, logits);

    dim3 grid2((BATCH + 255) / 256);
    moe_sigmoid_mean_kernel<<<grid2, 256, 0, stream>>>(logits, out);
}